// BlocksCore_76347338653989
// MI455X (gfx1250) — compile-verified
//
#include <hip/hip_runtime.h>
#include <math.h>

#define BSZ 256
#define LQ  256
#define NK  64
#define BH  512
#define SUBIN 1537

typedef float v2f __attribute__((ext_vector_type(2)));
typedef float v8f __attribute__((ext_vector_type(8)));

// ---------------- workspace layout (floats) ----------------
#define OFF_CC 0L
#define OFF_QQ (OFF_CC + (long)BSZ*2*NK)
#define OFF_S  (OFF_QQ + (long)BSZ*2*LQ)            // Sclip, overwritten in-place by S1
#define OFF_S2 (OFF_S  + (long)BSZ*2*NK*LQ)
#define OFF_T  (OFF_S2 + (long)BSZ*2*NK*LQ)
#define OFF_A  (OFF_T  + (long)BSZ*2*NK*NK)
#define OFF_B  (OFF_A  + (long)BSZ*2*NK*BH)
#define OFF_H  (OFF_B  + (long)BSZ*2*NK*BH)

// ---------------- WMMA helpers (FP32 16x16x4) ----------------
__device__ __forceinline__ v8f wmma4(v8f acc, v2f a, v2f b) {
  return __builtin_amdgcn_wmma_f32_16x16x4_f32(false, a, false, b, (short)0, acc,
                                               false, false);
}

// A stored [m][k] stride LDA (base at tile m=0,k=0).
// B: if BT, stored [n][k] stride LDB; else [k][n] stride LDB.
template<int LDA, int LDB, bool BT, int KC>
__device__ __forceinline__ void gemm_chunk(v8f& acc, const float* __restrict__ As,
                                           const float* __restrict__ Bs, int lane) {
  int half = lane >> 4, mn = lane & 15;
#pragma unroll
  for (int k = 0; k < KC; k += 4) {
    int kb = k + 2 * half;
    v2f a, b;
    a.x = As[mn * LDA + kb];
    a.y = As[mn * LDA + kb + 1];
    if (BT) { b.x = Bs[mn * LDB + kb];  b.y = Bs[mn * LDB + kb + 1]; }
    else    { b.x = Bs[kb * LDB + mn];  b.y = Bs[(kb + 1) * LDB + mn]; }
    acc = wmma4(acc, a, b);
  }
}

__device__ __forceinline__ void store_tile(v8f acc, float* dst, long rstride, int lane) {
  int half = lane >> 4, n = lane & 15;
#pragma unroll
  for (int v = 0; v < 8; ++v) dst[(long)(v + 8 * half) * rstride + n] = acc[v];
}

// ---------------- CDNA5 async global->LDS copy path ----------------
// GLOBAL_LOAD_ASYNC_TO_LDS_B128: per-lane 16B DMA, bypasses VGPRs, ASYNCcnt.
// LDS byte address = low 32 bits of the flat shared-pointer (ISA 10.2: LDS
// aperture keeps the LDS offset in addr[31:0]).
__device__ __forceinline__ void async_ld16(float* lds_dst, const float* g_src) {
  unsigned l = (unsigned)(uintptr_t)lds_dst;
  asm volatile("global_load_async_to_lds_b128 %0, %1, off"
               :: "v"(l), "v"(g_src) : "memory");
}
__device__ __forceinline__ void async_wait() {
  asm volatile("s_wait_asynccnt 0x0" ::: "memory");
}

// ---------------- LDS staging (256 threads) ----------------
__device__ __forceinline__ void stage64x64_async(float* dst, const float* src, long sstride) {
  int t = threadIdx.x, r = t >> 2, c0 = (t & 3) * 16;
  const float* s = src + (long)r * sstride + c0;
  float* d = dst + r * 68 + c0;
#pragma unroll
  for (int i = 0; i < 4; ++i) async_ld16(d + i * 4, s + i * 4);
}

__device__ __forceinline__ void stage64x128_async(float* dst, const float* src, long sstride) {
  int t = threadIdx.x, r = t >> 2, c0 = (t & 3) * 32;
  const float* s = src + (long)r * sstride + c0;
  float* d = dst + r * 132 + c0;
#pragma unroll
  for (int i = 0; i < 8; ++i) async_ld16(d + i * 4, s + i * 4);
}

// VGPR staging kept only where data is transformed in flight.
__device__ __forceinline__ void stage64x64_mul(float* dst, const float* src, long sstride,
                                               const float* w) {
  int t = threadIdx.x, r = t >> 2, c0 = (t & 3) * 16;
  const float* s = src + (long)r * sstride + c0;
  float* d = dst + r * 68 + c0;
#pragma unroll
  for (int i = 0; i < 4; ++i) {
    float4 v = *(const float4*)(s + i * 4);
    float4 wv = *(const float4*)(w + c0 + i * 4);
    d[i*4+0] = v.x*wv.x; d[i*4+1] = v.y*wv.y; d[i*4+2] = v.z*wv.z; d[i*4+3] = v.w*wv.w;
  }
}

// ================= K0: cC[c]=C.w4C, qQ[q]=Q.w4Q =================
__global__ __launch_bounds__(256) void k0_rowdots(
    float* __restrict__ ws, const float* __restrict__ node,
    const float* __restrict__ obs, const float* __restrict__ act,
    const float* __restrict__ w4C_o, const float* __restrict__ w4Q_o,
    const float* __restrict__ w4C_a, const float* __restrict__ w4Q_a) {
  int b = blockIdx.x, h = blockIdx.y;
  const float* Q = h ? act : obs;
  const float* wC = h ? w4C_a : w4C_o;
  const float* wQ = h ? w4Q_a : w4Q_o;
  int wave = threadIdx.x >> 5, lane = threadIdx.x & 31;
  long bh = (long)b * 2 + h;
  for (int r = wave; r < NK + LQ; r += 8) {
    const float *src, *w;
    if (r < NK) { src = node + ((long)b * NK + r) * BH; w = wC; }
    else        { src = Q + ((long)b * LQ + (r - NK)) * BH; w = wQ; }
    float s = 0.f;
    for (int d = lane; d < BH; d += 32) s += src[d] * w[d];
    for (int off = 16; off; off >>= 1) s += __shfl_xor(s, off, 32);
    if (lane == 0) {
      if (r < NK) ws[OFF_CC + bh * NK + r] = s;
      else        ws[OFF_QQ + bh * LQ + (r - NK)] = s;
    }
  }
}

// ============ K1: Sclip = clip(Cw@Q^T + cC + qQ + bias) ============
__global__ __launch_bounds__(256) void k1_scores(
    float* __restrict__ ws, const float* __restrict__ node,
    const float* __restrict__ obs, const float* __restrict__ act,
    const float* __restrict__ wmlu_o, const float* __restrict__ wmlu_a,
    const float* __restrict__ bias_o, const float* __restrict__ bias_a) {
  __shared__ __attribute__((aligned(16))) float Cw[64 * 68];
  __shared__ __attribute__((aligned(16))) float Qs[64 * 68];
  int b = blockIdx.x, h = blockIdx.y, q0 = blockIdx.z * 64;
  const float* Q   = h ? act : obs;
  const float* wml = h ? wmlu_a : wmlu_o;
  float bias = h ? bias_a[0] : bias_o[0];
  long bh = (long)b * 2 + h;
  int wave = threadIdx.x >> 5, lane = threadIdx.x & 31;
  v8f z = {0,0,0,0,0,0,0,0};
  v8f acc[2]; acc[0] = z; acc[1] = z;
  for (int kk = 0; kk < BH; kk += 64) {
    __syncthreads();
    stage64x64_async(Qs, Q + ((long)b * LQ + q0) * BH + kk, BH);
    stage64x64_mul(Cw, node + (long)b * NK * BH + kk, BH, wml + kk);
    async_wait();
    __syncthreads();
#pragma unroll
    for (int i = 0; i < 2; ++i) {
      int t = wave * 2 + i, tm = t >> 2, tn = t & 3;
      gemm_chunk<68, 68, true, 64>(acc[i], Cw + tm * 16 * 68, Qs + tn * 16 * 68, lane);
    }
  }
  const float* cC = ws + OFF_CC + bh * NK;
  const float* qQ = ws + OFF_QQ + bh * LQ;
  float* S = ws + OFF_S + bh * NK * LQ;
  int half = lane >> 4, n = lane & 15;
#pragma unroll
  for (int i = 0; i < 2; ++i) {
    int t = wave * 2 + i, tm = t >> 2, tn = t & 3;
#pragma unroll
    for (int v = 0; v < 8; ++v) {
      int c = tm * 16 + v + 8 * half;
      int q = q0 + tn * 16 + n;
      float s = acc[i][v] + cC[c] + qQ[q] + bias;
      s = fminf(15.f, fmaxf(-15.f, s));
      S[(long)c * LQ + q] = s;
    }
  }
}

// ============ K2: S2 = masked softmax over c (column) ============
__global__ __launch_bounds__(256) void k2_colsoftmax(float* __restrict__ ws,
                                                     const float* __restrict__ nmask) {
  int b = blockIdx.x, h = blockIdx.y, q = threadIdx.x;
  long bh = (long)b * 2 + h;
  const float* S = ws + OFF_S + bh * NK * LQ;
  float* S2 = ws + OFF_S2 + bh * NK * LQ;
  const float* cm = nmask + (long)b * NK;
  float xv[NK];
  float mx = -1e30f;
#pragma unroll
  for (int c = 0; c < NK; ++c) {
    float x = S[(long)c * LQ + q] * cm[c];
    xv[c] = x;
    mx = fmaxf(mx, x);
  }
  float sum = 0.f;
#pragma unroll
  for (int c = 0; c < NK; ++c) {
    float e = __expf(xv[c] - mx) * cm[c];
    xv[c] = e; sum += e;
  }
  float inv = 1.f / (sum + 1e-6f);
#pragma unroll
  for (int c = 0; c < NK; ++c) S2[(long)c * LQ + q] = xv[c] * inv;
}

// ======= K3: S1 = masked softmax over q (row), in place on Sclip =======
__global__ __launch_bounds__(256) void k3_rowsoftmax(float* __restrict__ ws,
                                                     const float* __restrict__ omask,
                                                     const float* __restrict__ amask) {
  int b = blockIdx.x, h = blockIdx.y;
  const float* qm = (h ? amask : omask) + (long)b * LQ;
  int wave = threadIdx.x >> 5, lane = threadIdx.x & 31;
  float* S = ws + OFF_S + ((long)b * 2 + h) * NK * LQ;
  for (int c = wave; c < NK; c += 8) {
    float x[8]; float mx = -1e30f;
#pragma unroll
    for (int i = 0; i < 8; ++i) {
      int q = lane + 32 * i;
      float v = S[(long)c * LQ + q] * qm[q];
      x[i] = v; mx = fmaxf(mx, v);
    }
    for (int off = 16; off; off >>= 1) mx = fmaxf(mx, __shfl_xor(mx, off, 32));
    float sum = 0.f;
#pragma unroll
    for (int i = 0; i < 8; ++i) {
      int q = lane + 32 * i;
      float e = __expf(x[i] - mx) * qm[q];
      x[i] = e; sum += e;
    }
    for (int off = 16; off; off >>= 1) sum += __shfl_xor(sum, off, 32);
    float inv = 1.f / (sum + 1e-6f);
#pragma unroll
    for (int i = 0; i < 8; ++i) S[(long)c * LQ + lane + 32 * i] = x[i] * inv;
  }
}

// ================= K4: T = S1 @ S2^T =================
__global__ __launch_bounds__(256) void k4_T(float* __restrict__ ws) {
  __shared__ __attribute__((aligned(16))) float S1c[64 * 68];
  __shared__ __attribute__((aligned(16))) float S2c[64 * 68];
  int b = blockIdx.x, h = blockIdx.y;
  long bh = (long)b * 2 + h;
  int wave = threadIdx.x >> 5, lane = threadIdx.x & 31;
  v8f z = {0,0,0,0,0,0,0,0};
  v8f acc[2]; acc[0] = z; acc[1] = z;
  const float* S1 = ws + OFF_S + bh * NK * LQ;
  const float* S2 = ws + OFF_S2 + bh * NK * LQ;
  for (int kk = 0; kk < LQ; kk += 64) {
    __syncthreads();
    stage64x64_async(S1c, S1 + kk, LQ);
    stage64x64_async(S2c, S2 + kk, LQ);
    async_wait();
    __syncthreads();
#pragma unroll
    for (int i = 0; i < 2; ++i) {
      int t = wave * 2 + i, tm = t >> 2, tn = t & 3;
      gemm_chunk<68, 68, true, 64>(acc[i], S1c + tm * 16 * 68, S2c + tn * 16 * 68, lane);
    }
  }
  float* T = ws + OFF_T + bh * NK * NK;
#pragma unroll
  for (int i = 0; i < 2; ++i) {
    int t = wave * 2 + i, tm = t >> 2, tn = t & 3;
    store_tile(acc[i], T + (long)(tm * 16) * NK + tn * 16, NK, lane);
  }
}

// ================= K5: A = S1 @ Q =================
__global__ __launch_bounds__(256) void k5_A(float* __restrict__ ws,
                                            const float* __restrict__ obs,
                                            const float* __restrict__ act) {
  __shared__ __attribute__((aligned(16))) float S1c[64 * 68];
  __shared__ __attribute__((aligned(16))) float Qc[64 * 132];
  int b = blockIdx.x, h = blockIdx.y, n0 = blockIdx.z * 128;
  long bh = (long)b * 2 + h;
  const float* Q = h ? act : obs;
  const float* S1 = ws + OFF_S + bh * NK * LQ;
  int wave = threadIdx.x >> 5, lane = threadIdx.x & 31;
  v8f z = {0,0,0,0,0,0,0,0};
  v8f acc[4]; acc[0] = z; acc[1] = z; acc[2] = z; acc[3] = z;
  for (int kk = 0; kk < LQ; kk += 64) {
    __syncthreads();
    stage64x64_async(S1c, S1 + kk, LQ);
    stage64x128_async(Qc, Q + ((long)b * LQ + kk) * BH + n0, BH);
    async_wait();
    __syncthreads();
#pragma unroll
    for (int i = 0; i < 4; ++i) {
      int t = wave * 4 + i, tm = t >> 3, tn = t & 7;
      gemm_chunk<68, 132, false, 64>(acc[i], S1c + tm * 16 * 68, Qc + tn * 16, lane);
    }
  }
  float* A = ws + OFF_A + bh * NK * BH;
#pragma unroll
  for (int i = 0; i < 4; ++i) {
    int t = wave * 4 + i, tm = t >> 3, tn = t & 7;
    store_tile(acc[i], A + (long)(tm * 16) * BH + n0 + tn * 16, BH, lane);
  }
}

// ================= K6: B = T @ C =================
__global__ __launch_bounds__(256) void k6_B(float* __restrict__ ws,
                                            const float* __restrict__ node) {
  __shared__ __attribute__((aligned(16))) float Tc[64 * 68];
  __shared__ __attribute__((aligned(16))) float Cc[64 * 132];
  int b = blockIdx.x, h = blockIdx.y, n0 = blockIdx.z * 128;
  long bh = (long)b * 2 + h;
  int wave = threadIdx.x >> 5, lane = threadIdx.x & 31;
  stage64x64_async(Tc, ws + OFF_T + bh * NK * NK, NK);
  stage64x128_async(Cc, node + (long)b * NK * BH + n0, BH);
  async_wait();
  __syncthreads();
  v8f z = {0,0,0,0,0,0,0,0};
  float* Bm = ws + OFF_B + bh * NK * BH;
#pragma unroll
  for (int i = 0; i < 4; ++i) {
    int t = wave * 4 + i, tm = t >> 3, tn = t & 7;
    v8f acc = z;
    gemm_chunk<68, 132, false, 64>(acc, Tc + tm * 16 * 68, Cc + tn * 16, lane);
    store_tile(acc, Bm + (long)(tm * 16) * BH + n0 + tn * 16, BH, lane);
  }
}

// ===== K7: h = C@P0 + A@P1 + (C.A)@P2 + (C.B)@P3 =====
__global__ __launch_bounds__(256) void k7_h(float* __restrict__ ws,
                                            const float* __restrict__ node,
                                            const float* __restrict__ prj_o,
                                            const float* __restrict__ prj_a) {
  __shared__ __attribute__((aligned(16))) float Fc[64 * 68];
  __shared__ __attribute__((aligned(16))) float Pc[64 * 132];
  int b = blockIdx.x, h = blockIdx.y, n0 = blockIdx.z * 128;
  long bh = (long)b * 2 + h;
  const float* prj = h ? prj_a : prj_o;
  const float* Cb = node + (long)b * NK * BH;
  const float* Ab = ws + OFF_A + bh * NK * BH;
  const float* Bb = ws + OFF_B + bh * NK * BH;
  int wave = threadIdx.x >> 5, lane = threadIdx.x & 31;
  v8f z = {0,0,0,0,0,0,0,0};
  v8f acc[4]; acc[0] = z; acc[1] = z; acc[2] = z; acc[3] = z;
  for (int tau = 0; tau < 4; ++tau) {
    for (int kk = 0; kk < BH; kk += 64) {
      __syncthreads();
      stage64x128_async(Pc, prj + ((long)tau * BH + kk) * BH + n0, BH);
      {
        int t = threadIdx.x, r = t >> 2, c0 = (t & 3) * 16;
        const float* Cp = Cb + (long)r * BH + kk + c0;
        const float* Ap = Ab + (long)r * BH + kk + c0;
        const float* Bp = Bb + (long)r * BH + kk + c0;
        float* d = Fc + r * 68 + c0;
#pragma unroll
        for (int i = 0; i < 4; ++i) {
          float4 cv = *(const float4*)(Cp + i * 4);
          float4 ov;
          if (tau == 0) ov = cv;
          else if (tau == 1) ov = *(const float4*)(Ap + i * 4);
          else if (tau == 2) {
            float4 av = *(const float4*)(Ap + i * 4);
            ov.x = cv.x * av.x; ov.y = cv.y * av.y; ov.z = cv.z * av.z; ov.w = cv.w * av.w;
          } else {
            float4 bv = *(const float4*)(Bp + i * 4);
            ov.x = cv.x * bv.x; ov.y = cv.y * bv.y; ov.z = cv.z * bv.z; ov.w = cv.w * bv.w;
          }
          d[i*4+0] = ov.x; d[i*4+1] = ov.y; d[i*4+2] = ov.z; d[i*4+3] = ov.w;
        }
      }
      async_wait();
      __syncthreads();
#pragma unroll
      for (int i = 0; i < 4; ++i) {
        int t = wave * 4 + i, tm = t >> 3, tn = t & 7;
        gemm_chunk<68, 132, false, 64>(acc[i], Fc + tm * 16 * 68, Pc + tn * 16, lane);
      }
    }
  }
  float* H = ws + OFF_H + bh * NK * BH;
#pragma unroll
  for (int i = 0; i < 4; ++i) {
    int t = wave * 4 + i, tm = t >> 3, tn = t & 7;
    store_tile(acc[i], H + (long)(tm * 16) * BH + n0 + tn * 16, BH, lane);
  }
}

// ===== K8: block-diagonal grouped linear + reward column + bias =====
__global__ __launch_bounds__(256) void k8_blk(float* __restrict__ out,
                                              const float* __restrict__ ws,
                                              const float* __restrict__ node,
                                              const float* __restrict__ blkW,
                                              const float* __restrict__ blkb,
                                              const float* __restrict__ rew) {
  __shared__ __attribute__((aligned(16))) float Xc[64 * 68];
  __shared__ __attribute__((aligned(16))) float Wc[64 * 132];
  int kn = blockIdx.x;           // node (block) index
  int b0 = blockIdx.y * 64;      // batch tile
  int n0 = blockIdx.z * 128;     // output tile
  int wave = threadIdx.x >> 5, lane = threadIdx.x & 31;
  v8f z = {0,0,0,0,0,0,0,0};
  v8f acc[4]; acc[0] = z; acc[1] = z; acc[2] = z; acc[3] = z;
  for (int tau = 0; tau < 3; ++tau) {
    const float* src; long sstride;
    if (tau < 2) { src = ws + OFF_H + (long)tau * NK * BH + (long)kn * BH; sstride = (long)2 * NK * BH; }
    else         { src = node + (long)kn * BH;                              sstride = (long)NK * BH; }
    for (int kk = 0; kk < BH; kk += 64) {
      __syncthreads();
      stage64x64_async(Xc, src + (long)b0 * sstride + kk, sstride);
      stage64x128_async(Wc, blkW + ((long)kn * SUBIN + tau * BH + kk) * BH + n0, BH);
      async_wait();
      __syncthreads();
#pragma unroll
      for (int i = 0; i < 4; ++i) {
        int t = wave * 4 + i, tm = t >> 3, tn = t & 7;
        gemm_chunk<68, 132, false, 64>(acc[i], Xc + tm * 16 * 68, Wc + tn * 16, lane);
      }
    }
  }
  const float* bb = blkb + (long)kn * BH + n0;
  const float* wl = blkW + ((long)kn * SUBIN + (SUBIN - 1)) * BH + n0;
  int half = lane >> 4, n = lane & 15;
#pragma unroll
  for (int i = 0; i < 4; ++i) {
    int t = wave * 4 + i, tm = t >> 3, tn = t & 7;
#pragma unroll
    for (int v = 0; v < 8; ++v) {
      int m = v + 8 * half;
      int bi = b0 + tm * 16 + m;
      int o = tn * 16 + n;
      out[((long)bi * NK + kn) * BH + n0 + o] = acc[i][v] + bb[o] + rew[bi] * wl[o];
    }
  }
}

// ================= launch =================
extern "C" void kernel_launch(void* const* d_in, const int* in_sizes, int n_in,
                              void* d_out, int out_size, void* d_ws, size_t ws_size,
                              hipStream_t stream) {
  (void)in_sizes; (void)n_in; (void)out_size; (void)ws_size;
  const float* act    = (const float*)d_in[0];
  const float* obs    = (const float*)d_in[1];
  const float* amask  = (const float*)d_in[2];
  const float* omask  = (const float*)d_in[3];
  const float* rew    = (const float*)d_in[4];
  const float* node   = (const float*)d_in[5];
  const float* nmask  = (const float*)d_in[6];
  const float* w4C_o  = (const float*)d_in[7];
  const float* w4Q_o  = (const float*)d_in[8];
  const float* wmlu_o = (const float*)d_in[9];
  const float* bias_o = (const float*)d_in[10];
  const float* w4C_a  = (const float*)d_in[11];
  const float* w4Q_a  = (const float*)d_in[12];
  const float* wmlu_a = (const float*)d_in[13];
  const float* bias_a = (const float*)d_in[14];
  const float* prj_o  = (const float*)d_in[15];
  const float* prj_a  = (const float*)d_in[16];
  const float* blkW   = (const float*)d_in[17];
  const float* blkb   = (const float*)d_in[18];
  float* out = (float*)d_out;
  float* ws  = (float*)d_ws;

  dim3 blk(256);
  k0_rowdots   <<<dim3(BSZ, 2),    blk, 0, stream>>>(ws, node, obs, act, w4C_o, w4Q_o, w4C_a, w4Q_a);
  k1_scores    <<<dim3(BSZ, 2, 4), blk, 0, stream>>>(ws, node, obs, act, wmlu_o, wmlu_a, bias_o, bias_a);
  k2_colsoftmax<<<dim3(BSZ, 2),    blk, 0, stream>>>(ws, nmask);
  k3_rowsoftmax<<<dim3(BSZ, 2),    blk, 0, stream>>>(ws, omask, amask);
  k4_T         <<<dim3(BSZ, 2),    blk, 0, stream>>>(ws);
  k5_A         <<<dim3(BSZ, 2, 4), blk, 0, stream>>>(ws, obs, act);
  k6_B         <<<dim3(BSZ, 2, 4), blk, 0, stream>>>(ws, node);
  k7_h         <<<dim3(BSZ, 2, 4), blk, 0, stream>>>(ws, node, prj_o, prj_a);
  k8_blk       <<<dim3(NK, 4, 4),  blk, 0, stream>>>(out, ws, node, blkW, blkb, rew);
}